// MoogVCFRFModelCell_55035710931668
// MI455X (gfx1250) — compile-verified
//
#include <hip/hip_runtime.h>

typedef float v2f __attribute__((ext_vector_type(2)));
typedef float v8f __attribute__((ext_vector_type(8)));

#define SR_F     44100.0f
#define R_PARAM  0.5f
#define F_PARAM  1000.0f

// Fold the 3 scalar params into M_total = Minv @ Mfwd, va = Minv@b_a, vb = Minv@b_1ma.
// Fully unrolled 4x4 Gauss-Jordan (matrix is diagonally dominant; no pivoting needed).
__device__ __forceinline__ void build_params(float gf, float gr, float alpha,
                                             float Mt[4][4], float va[4], float vb[4]) {
    const float k = 1.0f / SR_F;
    const float w = 2.0f * 3.14159265358979323846f * gf * F_PARAM;
    const float A[4][4] = {
        { -w, 0.f, 0.f, -4.0f * w * gr * R_PARAM },
        {  w,  -w, 0.f, 0.f },
        { 0.f,  w,  -w, 0.f },
        { 0.f, 0.f,  w,  -w }};
    float Mm[4][4], Ff[4][4], inv[4][4];
#pragma unroll
    for (int i = 0; i < 4; i++)
#pragma unroll
        for (int j = 0; j < 4; j++) {
            const float ka2 = 0.5f * k * A[i][j];
            const float id  = (i == j) ? 1.0f : 0.0f;
            Mm[i][j]  = id - ka2 * alpha;             // I - k*A/2*alpha
            Ff[i][j]  = id + k * A[i][j] - ka2 * alpha; // I + k*A - k*A/2*alpha
            inv[i][j] = id;
        }
#pragma unroll
    for (int p = 0; p < 4; p++) {
        const float r = 1.0f / Mm[p][p];
#pragma unroll
        for (int j = 0; j < 4; j++) { Mm[p][j] *= r; inv[p][j] *= r; }
#pragma unroll
        for (int i = 0; i < 4; i++) {
            if (i == p) continue;
            const float f = Mm[i][p];
#pragma unroll
            for (int j = 0; j < 4; j++) { Mm[i][j] -= f * Mm[p][j]; inv[i][j] -= f * inv[p][j]; }
        }
    }
#pragma unroll
    for (int i = 0; i < 4; i++)
#pragma unroll
        for (int j = 0; j < 4; j++) {
            float s = 0.0f;
#pragma unroll
            for (int kk = 0; kk < 4; kk++) s = fmaf(inv[i][kk], Ff[kk][j], s);
            Mt[i][j] = s;
        }
    const float kb2 = 0.5f * k * w;                 // first element of k*B/2
    const float ba  = kb2 * alpha;
    const float b1  = kb2 * (1.0f - 0.5f * alpha);
#pragma unroll
    for (int i = 0; i < 4; i++) { va[i] = inv[i][0] * ba; vb[i] = inv[i][0] * b1; }
}

__global__ __launch_bounds__(256)
void moog_wmma_kernel(const float* __restrict__ u_n,
                      const float* __restrict__ x_n1,
                      const float* __restrict__ u_n1,
                      const float* __restrict__ gf_p,
                      const float* __restrict__ gr_p,
                      const float* __restrict__ gain_p,
                      const float* __restrict__ alpha_p,
                      float* __restrict__ out, int B, int ntiles) {
    const int lane = threadIdx.x & 31;
    const int col  = lane & 15;   // sample column within 16-sample tile
    const int hi   = lane >> 4;   // 0: K={0,1}, 1: K={2,3} halves of A/B operands

    const float gain = *gain_p;
    float Mt[4][4], va[4], vb[4];
    build_params(*gf_p, *gr_p, *alpha_p, Mt, va, vb);

    // A operand of V_WMMA_F32_16X16X4_F32 (16x4, rows 4..15 zero):
    // lanes 0-15 hold row=lane K={0,1}; lanes 16-31 hold row=lane-16 K={2,3}.
    v2f a; a.x = 0.0f; a.y = 0.0f;
    if      (col == 0) { a.x = hi ? Mt[0][2] : Mt[0][0]; a.y = hi ? Mt[0][3] : Mt[0][1]; }
    else if (col == 1) { a.x = hi ? Mt[1][2] : Mt[1][0]; a.y = hi ? Mt[1][3] : Mt[1][1]; }
    else if (col == 2) { a.x = hi ? Mt[2][2] : Mt[2][0]; a.y = hi ? Mt[2][3] : Mt[2][1]; }
    else if (col == 3) { a.x = hi ? Mt[3][2] : Mt[3][0]; a.y = hi ? Mt[3][3] : Mt[3][1]; }

    float* __restrict__ yout = out;                  // [B]
    float* __restrict__ xout = out + (size_t)B;      // [B,4]
    float* __restrict__ uout = out + (size_t)5 * B;  // [B]

    const int nwaves = (int)((gridDim.x * blockDim.x) >> 5);
    const int wave   = (int)((blockIdx.x * blockDim.x + threadIdx.x) >> 5);

    for (int t = wave; t < ntiles; t += nwaves) {
        const int sc = (t << 4) + col;          // this lane's sample
        const int si = (sc < B) ? sc : (B - 1); // clamped index for loads

        // B operand (4x16): lane loads its sample's state pair -> coalesced b64.
        const v2f b = *(const v2f*)(x_n1 + ((size_t)si << 2) + (hi << 1));
        const float uu  = u_n[si];
        const float uu1 = u_n1[si];

        // C operand: input injection; rows 4..15 are don't-care.
        v8f c;
        c[0] = fmaf(va[0], uu, vb[0] * uu1);
        c[1] = fmaf(va[1], uu, vb[1] * uu1);
        c[2] = fmaf(va[2], uu, vb[2] * uu1);
        c[3] = fmaf(va[3], uu, vb[3] * uu1);
        c[4] = 0.0f; c[5] = 0.0f; c[6] = 0.0f; c[7] = 0.0f;

        // D[m][n] = sum_k Mt[m][k]*x_n1[k] + C[m][n]; 16 samples per instruction.
        v8f d = __builtin_amdgcn_wmma_f32_16x16x4_f32(
            /*neg_a=*/false, a, /*neg_b=*/false, b,
            /*c_mod=*/(short)0, c, /*reuse_a=*/false, /*reuse_b=*/false);

        // Lanes 0-15 hold sample=lane, states in d[0..3] -> coalesced b128 store.
        if (hi == 0 && sc < B) {
            *(float4*)(xout + ((size_t)sc << 2)) = make_float4(d[0], d[1], d[2], d[3]);
            yout[sc] = gain * d[3];
            uout[sc] = uu;
        }
    }
}

extern "C" void kernel_launch(void* const* d_in, const int* in_sizes, int n_in,
                              void* d_out, int out_size, void* d_ws, size_t ws_size,
                              hipStream_t stream) {
    (void)n_in; (void)out_size; (void)d_ws; (void)ws_size;
    const float* u_n   = (const float*)d_in[0];
    const float* x_n1  = (const float*)d_in[1];
    const float* u_n1  = (const float*)d_in[2];
    const float* gf    = (const float*)d_in[3];
    const float* gr    = (const float*)d_in[4];
    const float* gain  = (const float*)d_in[5];
    const float* alpha = (const float*)d_in[6];
    const int B = in_sizes[0];
    const int ntiles = (B + 15) >> 4;

    // 8 waves/block; aim for ~16 tiles per wave of grid-stride work.
    int blocks = (ntiles + 8 * 16 - 1) / (8 * 16);
    if (blocks < 1)    blocks = 1;
    if (blocks > 4096) blocks = 4096;

    moog_wmma_kernel<<<blocks, 256, 0, stream>>>(u_n, x_n1, u_n1, gf, gr, gain, alpha,
                                                 (float*)d_out, B, ntiles);
}